// OT_KD_55619826483665
// MI455X (gfx1250) — compile-verified
//
#include <hip/hip_runtime.h>
#include <hip/hip_bf16.h>
#include <math.h>

typedef __attribute__((ext_vector_type(2))) float v2f;
typedef __attribute__((ext_vector_type(8))) float v8f;

#define BATCH 64
#define T_CH  768
#define S_CH  640
#define HW    576   // 24*24
#define CC    64

// ---------------------------------------------------------------------------
// Kernel 1: spatial mean pooling (bandwidth-bound, ~208 MB traffic).
// Each wave reduces TWO adjacent channels (1152 floats = 288 float4 = 9/lane).
// ---------------------------------------------------------------------------
__global__ void ot_pool_kernel(const float* __restrict__ tf,
                               const float* __restrict__ sf,
                               float* __restrict__ tpool,
                               float* __restrict__ spool) {
  const int wid  = threadIdx.x >> 5;
  const int lane = threadIdx.x & 31;
  long p = (long)blockIdx.x * 8 + wid;               // channel-pair index
  const long TPAIRS = (long)BATCH * T_CH / 2;        // 24576
  const long SPAIRS = (long)BATCH * S_CH / 2;        // 20480
  const float4* base;
  float* out;
  if (p < TPAIRS) {
    base = (const float4*)(tf + p * (2 * HW));
    out  = tpool + p * 2;
  } else {
    long q = p - TPAIRS;
    if (q >= SPAIRS) return;
    base = (const float4*)(sf + q * (2 * HW));
    out  = spool + q * 2;
  }
  float acc0 = 0.f, acc1 = 0.f;
#pragma unroll
  for (int r = 0; r < 9; ++r) {
    int idx = r * 32 + lane;                         // float4 index in [0,288)
    float4 v = base[idx];
    float sum = v.x + v.y + v.z + v.w;
    if (idx < 144) acc0 += sum; else acc1 += sum;    // first 144 f4 = channel 0
  }
#pragma unroll
  for (int off = 16; off > 0; off >>= 1) {
    acc0 += __shfl_xor(acc0, off, 32);
    acc1 += __shfl_xor(acc1, off, 32);
  }
  if (lane == 0) {
    out[0] = acc0 * (1.0f / 576.0f);
    out[1] = acc1 * (1.0f / 576.0f);
  }
}

// ---------------------------------------------------------------------------
// Kernel 2: t = tpool @ Wt^T + bt and s = spool @ Ws^T + bs via
// V_WMMA_F32_16X16X4_F32 (full f32 precision, matches reference numerics).
// One wave (32 threads, EXEC all 1s) per 16x16 output tile; 32 tiles total.
// A layout (16x4 f32): lane = M (lanes 0-15 / 16-31 pick K pair), 2 VGPRs.
// B layout (4x16 f32): lane = N, mirrored K striping.
// C/D layout: col = lane%16, row = vgpr + 8*(lane/16).
// ---------------------------------------------------------------------------
__global__ void ot_gemm_wmma_kernel(const float* __restrict__ tpool,
                                    const float* __restrict__ spool,
                                    const float* __restrict__ Wt,
                                    const float* __restrict__ bt,
                                    const float* __restrict__ Ws,
                                    const float* __restrict__ bs,
                                    float* __restrict__ t,
                                    float* __restrict__ s) {
  const int wid   = blockIdx.x;       // 0..31
  const int which = wid >> 4;         // 0 = teacher, 1 = student
  const int tile  = wid & 15;
  const int mt = (tile >> 2) << 4;    // batch-row tile offset
  const int nt = (tile & 3) << 4;     // output-col tile offset
  const int K  = which ? S_CH : T_CH;
  const float* pool = which ? spool : tpool;  // [64, K]
  const float* W    = which ? Ws    : Wt;     // [CC, K] row-major
  const float* bias = which ? bs    : bt;
  float* out        = which ? s     : t;      // [64, CC]

  const int lane = threadIdx.x & 31;
  const int lo = lane & 15, hi = lane >> 4;
  const float* arow = pool + (long)(mt + lo) * K + hi * 2;
  const float* brow = W    + (long)(nt + lo) * K + hi * 2;

  v8f acc = {};
  for (int k = 0; k < K; k += 4) {
    v2f a = *(const v2f*)(arow + k);   // A[m=lo][k + 2*hi + {0,1}]
    v2f b = *(const v2f*)(brow + k);   // B[k + 2*hi + {0,1}][n=lo] = W[nt+lo][...]
    acc = __builtin_amdgcn_wmma_f32_16x16x4_f32(
        /*neg_a=*/false, a, /*neg_b=*/false, b,
        /*c_mod=*/(short)0, acc, /*reuse_a=*/false, /*reuse_b=*/false);
  }
  const float bn = bias[nt + lo];
#pragma unroll
  for (int v = 0; v < 8; ++v) {
    int m = v + 8 * hi;
    out[(mt + m) * CC + (nt + lo)] = acc[v] + bn;
  }
}

// ---------------------------------------------------------------------------
// Kernel 3: per-batch log-domain Sinkhorn (5 iters) + loss partials.
// One block (64 threads = 2 waves) per batch; M[i,j]=(t_i-s_j)^2 recomputed
// on the fly (reg = 1.0). Deterministic, no atomics.
// ---------------------------------------------------------------------------
__global__ void ot_sinkhorn_kernel(const float* __restrict__ t,
                                   const float* __restrict__ s,
                                   float* __restrict__ partials) {
  __shared__ float ts[64], ss[64], u[64], vv[64], red[2];
  const int b = blockIdx.x;
  const int i = threadIdx.x;  // 0..63
  ts[i] = t[b * 64 + i];
  ss[i] = s[b * 64 + i];
  __syncthreads();

  // log_softmax constants (redundant per-thread compute over 64 elems; trivial)
  float tmax = -INFINITY, smax = -INFINITY;
  for (int j = 0; j < 64; ++j) {
    tmax = fmaxf(tmax, ts[j]);
    smax = fmaxf(smax, ss[j]);
  }
  float tsum = 0.f, ssum = 0.f;
  for (int j = 0; j < 64; ++j) {
    tsum += expf(ts[j] - tmax);
    ssum += expf(ss[j] - smax);
  }
  const float loga = ts[i] - (tmax + logf(tsum));
  const float logb = ss[i] - (smax + logf(ssum));
  u[i] = 0.f; vv[i] = 0.f;
  __syncthreads();

  const float ti = ts[i], si = ss[i];
  for (int it = 0; it < 5; ++it) {
    // u_i = log_a_i - logsumexp_j( (t_i - s_j)^2 + v_j )
    float mx = -INFINITY;
    for (int j = 0; j < 64; ++j) { float d = ti - ss[j]; mx = fmaxf(mx, d * d + vv[j]); }
    float sm = 0.f;
    for (int j = 0; j < 64; ++j) { float d = ti - ss[j]; sm += expf(d * d + vv[j] - mx); }
    float unew = loga - (mx + logf(sm));
    __syncthreads();
    u[i] = unew;
    __syncthreads();
    // v_j = log_b_j - logsumexp_i( (t_i - s_j)^2 + u_i )   (thread i is col j=i)
    mx = -INFINITY;
    for (int k = 0; k < 64; ++k) { float d = ts[k] - si; mx = fmaxf(mx, d * d + u[k]); }
    sm = 0.f;
    for (int k = 0; k < 64; ++k) { float d = ts[k] - si; sm += expf(d * d + u[k] - mx); }
    float vnew = logb - (mx + logf(sm));
    __syncthreads();
    vv[i] = vnew;
    __syncthreads();
  }

  // mapped_j = sum_i exp(M[i,j] + u_i + v_j) * t_i ; partial = (mapped_j - s_j)^2
  const float vj = vv[i];
  float mapped = 0.f;
  for (int k = 0; k < 64; ++k) {
    float d = ts[k] - si;
    mapped += expf(d * d + u[k] + vj) * ts[k];
  }
  float diff = mapped - si;
  float part = diff * diff;
#pragma unroll
  for (int off = 16; off > 0; off >>= 1) part += __shfl_xor(part, off, 32);
  if ((i & 31) == 0) red[i >> 5] = part;
  __syncthreads();
  if (i == 0) partials[b] = red[0] + red[1];
}

// ---------------------------------------------------------------------------
// Kernel 4: final reduction of 64 per-batch partials -> mean over 64*64 elems.
// ---------------------------------------------------------------------------
__global__ void ot_finalize_kernel(const float* __restrict__ partials,
                                   float* __restrict__ out) {
  __shared__ float red[2];
  const int i = threadIdx.x;  // 64 threads
  float v = partials[i];
#pragma unroll
  for (int off = 16; off > 0; off >>= 1) v += __shfl_xor(v, off, 32);
  if ((i & 31) == 0) red[i >> 5] = v;
  __syncthreads();
  if (i == 0) out[0] = (red[0] + red[1]) * (1.0f / 4096.0f);
}

extern "C" void kernel_launch(void* const* d_in, const int* in_sizes, int n_in,
                              void* d_out, int out_size, void* d_ws, size_t ws_size,
                              hipStream_t stream) {
  const float* tf = (const float*)d_in[0];  // [64,768,24,24]
  const float* sf = (const float*)d_in[1];  // [64,640,24,24]
  const float* Wt = (const float*)d_in[2];  // [64,768]
  const float* bt = (const float*)d_in[3];  // [64]
  const float* Ws = (const float*)d_in[4];  // [64,640]
  const float* bs = (const float*)d_in[5];  // [64]

  float* ws       = (float*)d_ws;
  float* tpool    = ws;                         // 64*768 = 49152 floats
  float* spool    = tpool + BATCH * T_CH;       // 64*640 = 40960 floats
  float* t        = spool + BATCH * S_CH;       // 4096
  float* s        = t + BATCH * CC;             // 4096
  float* partials = s + BATCH * CC;             // 64

  // 45056 channel pairs / 8 pairs per block = 5632 blocks
  ot_pool_kernel<<<5632, 256, 0, stream>>>(tf, sf, tpool, spool);
  ot_gemm_wmma_kernel<<<32, 32, 0, stream>>>(tpool, spool, Wt, bt, Ws, bs, t, s);
  ot_sinkhorn_kernel<<<BATCH, 64, 0, stream>>>(t, s, partials);
  ot_finalize_kernel<<<1, 64, 0, stream>>>(partials, (float*)d_out);
}